// GCNDecor_953482740187
// MI455X (gfx1250) — compile-verified
//
#include <hip/hip_runtime.h>
#include <hip/hip_bf16.h>

#define GN 8192
#define GIN 128
#define GHID 64
#define GNC 2
#define GMLPW 64
#define GE 650000

#define KTILE 32                    // K columns staged per LDS tile
#define NTILES (GN / KTILE)         // 256
#define LDSROW 34                   // padded row stride (floats) -> conflict-free ds_load_b64
#define TILEFL (16 * LDSROW)        // 544 floats per buffer

typedef float v2f __attribute__((ext_vector_type(2)));
typedef float v8f __attribute__((ext_vector_type(8)));
typedef int v4i __attribute__((vector_size(16)));   // matches builtin's pointee type

// ---- CDNA5 async-copy helpers (ASYNCcnt-tracked global->LDS DMA) ----------
__device__ __forceinline__ void async_copy_b128(const float* g, float* l) {
#if __has_builtin(__builtin_amdgcn_global_load_async_to_lds_b128)
    __builtin_amdgcn_global_load_async_to_lds_b128(
        (__attribute__((address_space(1))) v4i*)g,
        (__attribute__((address_space(3))) v4i*)l, 0, 0);
#else
    unsigned loff = (unsigned)(size_t)(__attribute__((address_space(3))) void*)l;
    asm volatile("global_load_async_to_lds_b128 %0, %1, off"
                 :: "v"(loff), "v"(g) : "memory");
#endif
}

#if __has_builtin(__builtin_amdgcn_s_wait_asynccnt)
#define WAIT_ASYNC(n) __builtin_amdgcn_s_wait_asynccnt(n)
#else
#define WAIT_ASYNC(n) asm volatile("s_wait_asynccnt %0" :: "i"(n) : "memory")
#endif

// ---------------------------------------------------------------------------
// Column sums of adj (partial over 512-row segments, atomically combined).
// ---------------------------------------------------------------------------
__global__ __launch_bounds__(256) void colsum_part(const float* __restrict__ adj,
                                                   float* __restrict__ csum) {
    int col = blockIdx.x * 256 + threadIdx.x;
    const float* p = adj + (size_t)(blockIdx.y * 512) * GN + col;
    float s = 0.0f;
#pragma unroll 8
    for (int i = 0; i < 512; ++i) {
        s += __builtin_nontemporal_load(p);
        p += GN;
    }
    atomicAdd(&csum[col], s);
}

__global__ __launch_bounds__(256) void finish_dcol(const float* __restrict__ csum,
                                                   float* __restrict__ dcol) {
    int j = blockIdx.x * 256 + threadIdx.x;
    dcol[j] = rsqrtf(csum[j] + 1.0f);   // A = adj + I  => never inf
}

// ---------------------------------------------------------------------------
// Per-edge MLP gather/scatter. Weights cached in LDS.
// ---------------------------------------------------------------------------
__global__ __launch_bounds__(256) void edge_mlp(const int* __restrict__ idx,
                                                const float* __restrict__ adj,
                                                const float* __restrict__ xdeg,
                                                const float* __restrict__ ydeg,
                                                const float* __restrict__ Wm1,
                                                const float* __restrict__ bm1,
                                                const float* __restrict__ Wm2,
                                                const float* __restrict__ bm2,
                                                float* __restrict__ mask) {
    __shared__ float sm[386];
    int tid = threadIdx.x;
    if (tid < 192) sm[tid] = Wm1[tid];
    if (tid < 64) sm[192 + tid] = bm1[tid];
    if (tid < 128) sm[256 + tid] = Wm2[tid];
    if (tid < 2) sm[384 + tid] = bm2[tid];
    __syncthreads();
    int e = blockIdx.x * 256 + tid;
    if (e >= GE) return;
    int gi = idx[e];
    float a = adj[gi];
    float xd = xdeg[gi];
    float yd = ydeg[gi];
    float l0 = sm[384], l1 = sm[385];
#pragma unroll 8
    for (int j = 0; j < 64; ++j) {
        float h = fmaf(a, sm[j], fmaf(xd, sm[64 + j], fmaf(yd, sm[128 + j], sm[192 + j])));
        h = fmaxf(h, 0.0f);
        l0 = fmaf(h, sm[256 + 2 * j], l0);
        l1 = fmaf(h, sm[256 + 2 * j + 1], l1);
    }
    float mx = fmaxf(l0, l1);
    float e0 = __expf(l0 - mx), e1 = __expf(l1 - mx);
    mask[gi] = e1 / (e0 + e1);
}

// ---------------------------------------------------------------------------
// prep1: Bs1 = dcol ⊙ (x @ W1) in WMMA-B fragment pack layout (NT=4 tiles).
// ---------------------------------------------------------------------------
__global__ __launch_bounds__(256) void prep1(const float* __restrict__ x,
                                             const float* __restrict__ W1,
                                             const float* __restrict__ dcol,
                                             float* __restrict__ bpack) {
    int t = blockIdx.x * 256 + threadIdx.x;
    int row = t >> 6;
    int n = t & 63;
    const float* xr = x + (size_t)row * GIN;
    const float* w = W1 + n;
    float s = 0.0f;
#pragma unroll 8
    for (int q = 0; q < GIN; ++q) s = fmaf(xr[q], w[q * 64], s);
    s *= dcol[row];
    int k = row;
    int fidx = ((((k >> 2) * 4) + (n >> 4)) * 32 + ((k >> 1) & 1) * 16 + (n & 15)) * 2 + (k & 1);
    bpack[fidx] = s;
}

// ---------------------------------------------------------------------------
// prep2: Bs2 = dcol ⊙ (hid @ W2), padded to one 16-col tile (NT=1).
// ---------------------------------------------------------------------------
__global__ __launch_bounds__(256) void prep2(const float* __restrict__ hid,
                                             const float* __restrict__ W2,
                                             const float* __restrict__ dcol,
                                             float* __restrict__ bpack2) {
    int row = blockIdx.x * 256 + threadIdx.x;
    const float* h = hid + (size_t)row * GHID;
    float u0 = 0.0f, u1 = 0.0f;
#pragma unroll
    for (int j = 0; j < GHID; ++j) {
        float hv = h[j];
        u0 = fmaf(hv, W2[2 * j], u0);
        u1 = fmaf(hv, W2[2 * j + 1], u1);
    }
    float dc = dcol[row];
    u0 *= dc;
    u1 *= dc;
    int k = row;
    int base = ((k >> 2) * 32 + ((k >> 1) & 1) * 16) * 2 + (k & 1);
#pragma unroll
    for (int n = 0; n < 16; ++n) {
        float val = (n == 0) ? u0 : ((n == 1) ? u1 : 0.0f);
        bpack2[base + n * 2] = val;
    }
}

// ---------------------------------------------------------------------------
// Main streaming matmul: out = drow ⊙ ((adj + I) @ Bs)
//   - adj staged into per-wave LDS tiles via GLOBAL_LOAD_ASYNC_TO_LDS_B128
//     (ASYNCcnt double-buffered; 128B-contiguous DRAM transfers)
//   - A fragments read back with conflict-free ds_load_b64
//   - Bs pre-packed in WMMA-B fragment layout (L2-resident global b64 loads)
//   - fp32 WMMA 16x16x4; row sums accumulated for free -> drow = rsqrt(rs+1)
//   - "+I" handled by adding Bs[row] in the epilogue
// Block = 128 threads (4 waves); each wave owns a 16-row strip; grid = 128.
// ---------------------------------------------------------------------------
template <int NT, int MODE>
__global__ __launch_bounds__(128) void gcn_spmm(const float* __restrict__ adj,
                                                const float* __restrict__ bpack,
                                                float* __restrict__ outp) {
    __shared__ float tile[4][2][TILEFL];      // per-wave double buffers (17.4 KB)

    const int tid = threadIdx.x;
    const int lane = tid & 31;
    const int wave = tid >> 5;
    const int r0 = blockIdx.x * 64 + wave * 16;
    const int khalf = lane >> 4;              // 0 or 1
    const float* grow = adj + (size_t)r0 * GN;
    const v2f* bp = (const v2f*)bpack;

    // lane -> (row, col) inside the 16 x KTILE tile for the async fill
    const int fr = lane >> 3;                 // row subgroup 0..3 (per i: row i*4+fr)
    const int fc = (lane & 7) * 4;            // 16B-aligned column chunk

    v8f acc[NT];
#pragma unroll
    for (int c = 0; c < NT; ++c) {
        v8f z = {};
        acc[c] = z;
    }
    float rs = 0.0f;

    // issue tile kt into buffer kt&1 : 4 async b128 per wave, 128B/row coalesced
    auto issue = [&](int kt) {
        const float* gb = grow + kt * KTILE;
        float* lb = &tile[wave][kt & 1][0];
#pragma unroll
        for (int i = 0; i < 4; ++i) {
            int r = i * 4 + fr;
            const float* g = gb + (size_t)r * GN + fc;
            __builtin_prefetch(g + 2 * KTILE, 0, 0);   // warm L2 two tiles ahead
            async_copy_b128(g, lb + r * LDSROW + fc);
        }
    };

    issue(0);
    for (int kt = 0; kt < NTILES; ++kt) {
        if (kt + 1 < NTILES) {
            issue(kt + 1);
            WAIT_ASYNC(4);                    // current tile's 4 transfers done
        } else {
            WAIT_ASYNC(0);
        }
        const float* lt = &tile[wave][kt & 1][(lane & 15) * LDSROW + khalf * 2];
#pragma unroll
        for (int s = 0; s < KTILE / 4; ++s) {
            v2f a = *(const v2f*)(lt + s * 4);        // ds_load_b64 A fragment
            rs += a.x + a.y;
            int ks = kt * (KTILE / 4) + s;
#pragma unroll
            for (int c = 0; c < NT; ++c) {
                v2f b = bp[(ks * NT + c) * 32 + lane];
                acc[c] = __builtin_amdgcn_wmma_f32_16x16x4_f32(
                    false, a, false, b, (short)0, acc[c], false, false);
            }
        }
    }

    // Row sum of adj row (r0 + lane%16) is split across lanes L and L^16.
    float rowtot = rs + __shfl_xor(rs, 16, 32);
    float drow_my = rsqrtf(rowtot + 1.0f);    // +1 from the identity

#pragma unroll
    for (int v = 0; v < 8; ++v) {
        int m = khalf * 8 + v;                // local output row of acc elem v
        float drow_v = __shfl(drow_my, m, 32);
        int g = r0 + m;
#pragma unroll
        for (int c = 0; c < NT; ++c) {
            int n = lane & 15;
            int colg = c * 16 + n;
            int fidx = ((((g >> 2) * NT) + c) * 32 + ((g >> 1) & 1) * 16 + n) * 2 + (g & 1);
            float val = (acc[c][v] + bpack[fidx]) * drow_v;
            if (MODE == 0) {
                outp[(size_t)g * (NT * 16) + colg] = val;
            } else {
                if (colg < GNC) outp[(size_t)g * GNC + colg] = val;
            }
        }
    }
}

// ---------------------------------------------------------------------------
extern "C" void kernel_launch(void* const* d_in, const int* in_sizes, int n_in,
                              void* d_out, int out_size, void* d_ws, size_t ws_size,
                              hipStream_t stream) {
    const float* x    = (const float*)d_in[0];
    const float* adj  = (const float*)d_in[1];
    const float* xdeg = (const float*)d_in[2];
    const float* ydeg = (const float*)d_in[3];
    const float* Wm1  = (const float*)d_in[4];
    const float* bm1  = (const float*)d_in[5];
    const float* Wm2  = (const float*)d_in[6];
    const float* bm2  = (const float*)d_in[7];
    const float* W1   = (const float*)d_in[8];
    const float* W2   = (const float*)d_in[9];
    const int*   idx  = (const int*)d_in[10];

    float* out  = (float*)d_out;                          // [GN * GNC]
    float* mask = out + (size_t)GN * GNC;                 // [GN * GN]

    float* ws     = (float*)d_ws;
    float* csum   = ws;                                   // GN
    float* dcol   = ws + GN;                              // GN
    float* bpack1 = ws + 2 * GN;                          // 524288
    float* hid    = bpack1 + (GN / 4) * 4 * 64;           // 524288
    float* bpack2 = hid + (size_t)GN * GHID;              // 131072

    (void)hipMemsetAsync(mask, 0, (size_t)GN * GN * sizeof(float), stream);
    (void)hipMemsetAsync(csum, 0, GN * sizeof(float), stream);

    colsum_part<<<dim3(GN / 256, 16), 256, 0, stream>>>(adj, csum);
    finish_dcol<<<GN / 256, 256, 0, stream>>>(csum, dcol);

    edge_mlp<<<(GE + 255) / 256, 256, 0, stream>>>(idx, adj, xdeg, ydeg,
                                                   Wm1, bm1, Wm2, bm2, mask);

    prep1<<<(GN * GHID) / 256, 256, 0, stream>>>(x, W1, dcol, bpack1);
    gcn_spmm<4, 0><<<GN / 64, 128, 0, stream>>>(adj, bpack1, hid);

    prep2<<<GN / 256, 256, 0, stream>>>(hid, W2, dcol, bpack2);
    gcn_spmm<1, 1><<<GN / 64, 128, 0, stream>>>(adj, bpack2, out);
}